// GQAAttention_3118146257305
// MI455X (gfx1250) — compile-verified
//
#include <hip/hip_runtime.h>

// ---------------- problem constants ----------------
#define HID   1536
#define NH    12
#define NKV   2
#define NG    6
#define HD    128
#define BSZ   16
#define NBATCH 32
#define MBLK  256

typedef float v2f __attribute__((ext_vector_type(2)));
typedef float v4f __attribute__((ext_vector_type(4)));
typedef float v8f __attribute__((ext_vector_type(8)));
typedef unsigned int u32x4 __attribute__((ext_vector_type(4)));
typedef int i32x4 __attribute__((ext_vector_type(4)));
typedef int i32x8 __attribute__((ext_vector_type(8)));

// D = A(16x4) * B(4x16) + C  (fp32 WMMA, wave32)
static __device__ __forceinline__ v8f wmma4(v2f a, v2f b, v8f c) {
  return __builtin_amdgcn_wmma_f32_16x16x4_f32(false, a, false, b, (short)0, c,
                                               false, false);
}

// TDM: DMA a 16-row x 128-f32 tile (global row stride 256 f32) into LDS,
// inserting a 4-DWORD pad after every 128 DWORDs -> LDS row stride 132 f32
// (bank-conflict-free column reads for the QK^T B-fragments).
// D# packing per CDNA5 ISA.
static __device__ __forceinline__ void tdm_load_k_tile(const float* gsrc,
                                                       float* lds_dst) {
  const unsigned long long ga = (unsigned long long)(size_t)gsrc;
  u32x4 g0;
  g0.x = 1u;                                   // count=1, user descriptor
  g0.y = (unsigned)(size_t)(void*)lds_dst;     // lds_addr (bytes)
  g0.z = (unsigned)ga;                         // global_addr[31:0]
  g0.w = (unsigned)(ga >> 32) | 0x80000000u;   // global_addr[56:32] | type=2
  i32x8 g1;
  g1[0] = (int)((2u << 16)      // data_size = 4B
                | (1u << 20)    // pad_enable
                | (6u << 22)    // pad_interval: every 128 DWORDs
                | (3u << 25));  // pad_amount: 4 DWORDs
  g1[1] = (int)(128u << 16);    // tensor_dim0 = 128        (bits 79:48 lo)
  g1[2] = (int)(16u << 16);     // tensor_dim0 hi=0, tensor_dim1 = 16
  g1[3] = (int)(128u << 16);    // tensor_dim1 hi=0, tile_dim0 = 128
  g1[4] = 16;                   // tile_dim1 = 16, tile_dim2 = 0
  g1[5] = 256;                  // tensor_dim0_stride = 256 (f32 units)
  g1[6] = 0;                    // stride0 hi, tensor_dim1_stride lo
  g1[7] = 0;
  i32x4 gz = {0, 0, 0, 0};      // groups 2/3 unused (2D tile)
#if __clang_major__ >= 23
  i32x8 gz8 = {0, 0, 0, 0, 0, 0, 0, 0};
  __builtin_amdgcn_tensor_load_to_lds(g0, g1, gz, gz, gz8, 0);
#else
  __builtin_amdgcn_tensor_load_to_lds(g0, g1, gz, gz, 0);
#endif
}

// ============================================================
// Kernel 1: fused QKV projection.  Y[32,2048] = X[32,1536] @ Wcat^T + bcat
// One wave per 16-wide N tile; both M tiles (batches 0-15 / 16-31) so each
// weight row streams from HBM exactly once.
// ============================================================
__global__ __launch_bounds__(32) void qkv_proj_kernel(
    const float* __restrict__ x,
    const float* __restrict__ Wq, const float* __restrict__ bq,
    const float* __restrict__ Wk, const float* __restrict__ bk,
    const float* __restrict__ Wv, const float* __restrict__ bv,
    float* __restrict__ y) {
  const int n0   = blockIdx.x * 16;
  const int lane = threadIdx.x;
  const int half = lane >> 4, ln = lane & 15;

  const float* W; const float* bias; int wn;
  if (n0 < HID)            { W = Wq; bias = bq; wn = n0; }
  else if (n0 < HID + 256) { W = Wk; bias = bk; wn = n0 - HID; }
  else                     { W = Wv; bias = bv; wn = n0 - HID - 256; }

  const float* wr = W + (size_t)(wn + ln) * HID + 2 * half;   // B frag source
  const float* x0 = x + (size_t)ln * HID + 2 * half;          // batches 0-15
  const float* x1 = x + (size_t)(16 + ln) * HID + 2 * half;   // batches 16-31

  v8f acc0 = {}, acc1 = {};
#pragma unroll 8
  for (int k = 0; k < HID; k += 4) {
    v2f bf; bf.x = wr[k]; bf.y = wr[k + 1];
    v2f a0; a0.x = x0[k]; a0.y = x0[k + 1];
    v2f a1; a1.x = x1[k]; a1.y = x1[k + 1];
    acc0 = wmma4(a0, bf, acc0);
    acc1 = wmma4(a1, bf, acc1);
  }
  const float bb = bias[wn + ln];
#pragma unroll
  for (int i = 0; i < 8; i++) {
    y[(size_t)(i + 8 * half) * 2048 + n0 + ln]      = acc0[i] + bb;
    y[(size_t)(16 + i + 8 * half) * 2048 + n0 + ln] = acc1[i] + bb;
  }
}

// ============================================================
// Kernel 2: RoPE on q (12 heads) and new k (2 heads); v passthrough.
// ============================================================
__global__ __launch_bounds__(128) void rope_kernel(
    const float* __restrict__ y, const float* __restrict__ cosb,
    const float* __restrict__ sinb, float* __restrict__ qw,
    float* __restrict__ kw, float* __restrict__ vw) {
  const int b = blockIdx.x;
  const int d = threadIdx.x;
  const float c = cosb[b * HD + d];
  const float s = sinb[b * HD + d];
  const int dr = (d < 64) ? d + 64 : d - 64;
  const float sgn = (d < 64) ? -1.0f : 1.0f;
  const float* yb = y + (size_t)b * 2048;
#pragma unroll
  for (int h = 0; h < NH; h++) {
    float t = yb[h * HD + d];
    float r = yb[h * HD + dr];
    qw[((size_t)b * NH + h) * HD + d] = t * c + sgn * r * s;
  }
#pragma unroll
  for (int kv = 0; kv < NKV; kv++) {
    float t = yb[HID + kv * HD + d];
    float r = yb[HID + kv * HD + dr];
    kw[((size_t)b * NKV + kv) * HD + d] = t * c + sgn * r * s;
    vw[((size_t)b * NKV + kv) * HD + d] = yb[HID + 256 + kv * HD + d];
  }
}

// ============================================================
// Kernel 3: flash-decode GQA attention.  Block = (b, kv), 4 waves split the
// 16-key chunks; online softmax; fresh k/v substituted at position ctx-1
// (input caches are never written).  K tiles arrive via the Tensor Data
// Mover (tensor_load_to_lds + s_wait_tensorcnt); V tiles are register-staged
// during the DMA flight.  Matmuls via v_wmma_f32_16x16x4_f32.
// ============================================================
#define WAVE_LDS 2464                       // floats per wave region
#define ATTN_LDS_BYTES (4 * WAVE_LDS * 4)   // 39424 B dynamic LDS

__global__ __launch_bounds__(128) void attn_kernel(
    const float* __restrict__ qw, const float* __restrict__ kw,
    const float* __restrict__ vw, const float* __restrict__ k_cache,
    const float* __restrict__ v_cache, const int* __restrict__ block_table,
    const int* __restrict__ context_lens, float* __restrict__ attn_out) {
  extern __shared__ float smem[];
  const int b  = blockIdx.x >> 1;
  const int kv = blockIdx.x & 1;
  const int tid  = threadIdx.x;
  const int wave = tid >> 5, lane = tid & 31;
  const int half = lane >> 4, ln = lane & 15;

  float* wl   = smem + wave * WAVE_LDS;
  float* Kv   = wl;            // 16 x 132 tile (K via TDM, then V, then O)
  float* Pt   = wl + 2112;     // 16 x 20 score/prob tile (col 16 = rescale f)
  float* Mrow = wl + 2432;     // 16
  float* Lrow = wl + 2448;     // 16

  const int ctx = context_lens[b];
  const int pos = ctx - 1;
  const int pc = pos >> 4, ps = pos & 15;   // chunk/slot holding fresh k,v
  const int nchunks = (ctx + 15) >> 4;

  // ---- Q A-fragments (rows 0-5 real heads, 6-15 zero padding) ----
  const float* qbase = qw + ((size_t)b * NH + kv * NG) * HD;
  v2f qa[32];
#pragma unroll
  for (int ks = 0; ks < 32; ks++) {
    if (ln < NG) {
      qa[ks].x = qbase[ln * HD + ks * 4 + 2 * half];
      qa[ks].y = qbase[ln * HD + ks * 4 + 2 * half + 1];
    } else {
      qa[ks].x = 0.0f; qa[ks].y = 0.0f;
    }
  }

  v8f o[8];
#pragma unroll
  for (int t = 0; t < 8; t++) o[t] = (v8f){};
  float m = -1e30f, l = 0.0f;               // per-row stats (lanes 0-15)
  const float scale = 0.08838834764831845f; // 1/sqrt(128)

  const float* knew = kw + ((size_t)b * NKV + kv) * HD;
  const float* vnew = vw + ((size_t)b * NKV + kv) * HD;

  for (int c = wave; c < nchunks; c += 4) {
    const int blk = block_table[b * MBLK + c];
    const float* kb = k_cache + (size_t)blk * (BSZ * NKV * HD) + kv * HD;
    const float* vb = v_cache + (size_t)blk * (BSZ * NKV * HD) + kv * HD;

    // ---- kick off K tile DMA (TDM) into this wave's LDS buffer ----
    tdm_load_k_tile(kb, Kv);

    // prefetch next chunk's K rows (lanes 0-15) and V rows (lanes 16-31)
    if (c + 4 < nchunks) {
      const int nblk = block_table[b * MBLK + c + 4];
      const float* pb = (half ? v_cache : k_cache) +
                        (size_t)nblk * (BSZ * NKV * HD) + kv * HD +
                        (size_t)ln * (NKV * HD);
      __builtin_prefetch(pb, 0, 3);
    }

    // ---- V tile -> registers while the K DMA is in flight ----
    v4f vreg[16];
#pragma unroll
    for (int i = 0; i < 16; i++) {
      const bool ov = (c == pc) && (i == ps);
      const v4f* vsrc = (const v4f*)(ov ? vnew : vb + (size_t)i * (NKV * HD));
      vreg[i] = vsrc[lane];
    }

    // ---- wait for K tile, then patch in the fresh (RoPE'd) key row ----
    __builtin_amdgcn_s_wait_tensorcnt(0);
    if (c == pc)
      *(v4f*)(Kv + ps * 132 + lane * 4) = ((const v4f*)knew)[lane];

    // ---- scores S = Q * K^T : 32 k-steps of f32 WMMA ----
    v8f s = (v8f){};
#pragma unroll
    for (int ks = 0; ks < 32; ks++) {
      v2f bf;
      const int off = ln * 132 + ks * 4 + 2 * half; // B[k][n] = Kc[n][k]
      bf.x = Kv[off]; bf.y = Kv[off + 1];
      s = wmma4(qa[ks], bf, s);
    }

    // ---- scale + length mask, dump to LDS score tile ----
    const int j = c * 16 + ln;                 // this lane's key index (N=ln)
    const bool valid = j < ctx;
#pragma unroll
    for (int i = 0; i < 8; i++) {
      const float sv = valid ? s[i] * scale : -1e30f;
      Pt[(i + 8 * half) * 20 + ln] = sv;
    }

    // ---- V tile registers -> LDS (reuse Kv; per-wave DS ops are in-order) --
#pragma unroll
    for (int i = 0; i < 16; i++)
      *(v4f*)(Kv + i * 132 + lane * 4) = vreg[i];

    // ---- online softmax stats: lane ln owns row ln ----
    if (half == 0) {
      float rmax = -1e30f;
#pragma unroll
      for (int jj = 0; jj < 16; jj++) rmax = fmaxf(rmax, Pt[ln * 20 + jj]);
      const float mnew = fmaxf(m, rmax);
      const float f = expf(m - mnew);
      float sum = 0.0f;
#pragma unroll
      for (int jj = 0; jj < 16; jj++) {
        const float p = expf(Pt[ln * 20 + jj] - mnew);
        Pt[ln * 20 + jj] = p;
        sum += p;
      }
      l = l * f + sum;
      m = mnew;
      Pt[ln * 20 + 16] = f;                    // broadcast rescale factor
    }

    // ---- rescale accumulators by this chunk's correction factor ----
#pragma unroll
    for (int i = 0; i < 8; i++) {
      const float fi = Pt[(i + 8 * half) * 20 + 16];
#pragma unroll
      for (int t = 0; t < 8; t++) o[t][i] *= fi;
    }

    // ---- O += P(16x16) * V(16x128) : 8 d-tiles x 4 k-steps ----
    v2f pa[4];
#pragma unroll
    for (int ks = 0; ks < 4; ks++) {
      const int off = ln * 20 + ks * 4 + 2 * half;
      pa[ks].x = Pt[off]; pa[ks].y = Pt[off + 1];
    }
#pragma unroll
    for (int t = 0; t < 8; t++) {
#pragma unroll
      for (int ks = 0; ks < 4; ks++) {
        v2f bf;
        const int off = (ks * 4 + 2 * half) * 132 + t * 16 + ln;
        bf.x = Kv[off]; bf.y = Kv[off + 132];
        o[t] = wmma4(pa[ks], bf, o[t]);
      }
    }
  }

  // ---- stage per-wave partials into LDS ----
#pragma unroll
  for (int t = 0; t < 8; t++)
#pragma unroll
    for (int i = 0; i < 8; i++)
      Kv[(i + 8 * half) * 128 + t * 16 + ln] = o[t][i];
  if (half == 0) { Mrow[ln] = m; Lrow[ln] = l; }
  __syncthreads();

  // ---- cross-wave combine: thread d in 0..127, rows g in 0..5 ----
  const int d = tid;
#pragma unroll
  for (int g = 0; g < NG; g++) {
    float mstar = -1e30f;
#pragma unroll
    for (int w = 0; w < 4; w++)
      mstar = fmaxf(mstar, smem[w * WAVE_LDS + 2432 + g]);
    float num = 0.0f, den = 0.0f;
#pragma unroll
    for (int w = 0; w < 4; w++) {
      const float fw = expf(smem[w * WAVE_LDS + 2432 + g] - mstar);
      den += smem[w * WAVE_LDS + 2448 + g] * fw;
      num += smem[w * WAVE_LDS + g * 128 + d] * fw;
    }
    attn_out[((size_t)b * NH + kv * NG + g) * HD + d] = num / den;
  }
}

// ============================================================
// Kernel 4: output projection.  out[32,1536] = attn[32,1536] @ Wo^T
// ============================================================
__global__ __launch_bounds__(32) void out_proj_kernel(
    const float* __restrict__ attn, const float* __restrict__ Wo,
    float* __restrict__ out) {
  const int n0   = blockIdx.x * 16;
  const int lane = threadIdx.x;
  const int half = lane >> 4, ln = lane & 15;

  const float* wr = Wo + (size_t)(n0 + ln) * HID + 2 * half;
  const float* x0 = attn + (size_t)ln * HID + 2 * half;
  const float* x1 = attn + (size_t)(16 + ln) * HID + 2 * half;

  v8f acc0 = {}, acc1 = {};
#pragma unroll 8
  for (int k = 0; k < HID; k += 4) {
    v2f bf; bf.x = wr[k]; bf.y = wr[k + 1];
    v2f a0; a0.x = x0[k]; a0.y = x0[k + 1];
    v2f a1; a1.x = x1[k]; a1.y = x1[k + 1];
    acc0 = wmma4(a0, bf, acc0);
    acc1 = wmma4(a1, bf, acc1);
  }
#pragma unroll
  for (int i = 0; i < 8; i++) {
    out[(size_t)(i + 8 * half) * HID + n0 + ln]      = acc0[i];
    out[(size_t)(16 + i + 8 * half) * HID + n0 + ln] = acc1[i];
  }
}

// ============================================================
extern "C" void kernel_launch(void* const* d_in, const int* in_sizes, int n_in,
                              void* d_out, int out_size, void* d_ws,
                              size_t ws_size, hipStream_t stream) {
  (void)in_sizes; (void)n_in; (void)out_size; (void)ws_size;
  const float* x       = (const float*)d_in[0];
  const float* cosb    = (const float*)d_in[1];
  const float* sinb    = (const float*)d_in[2];
  const float* k_cache = (const float*)d_in[3];
  const float* v_cache = (const float*)d_in[4];
  const float* Wq      = (const float*)d_in[5];
  const float* bq      = (const float*)d_in[6];
  const float* Wk      = (const float*)d_in[7];
  const float* bk      = (const float*)d_in[8];
  const float* Wv      = (const float*)d_in[9];
  const float* bv      = (const float*)d_in[10];
  const float* Wo      = (const float*)d_in[11];
  const int* block_table  = (const int*)d_in[12];
  const int* context_lens = (const int*)d_in[13];

  float* ws   = (float*)d_ws;
  float* y    = ws;                  // 32*2048  qkv pre-rope
  float* qw   = ws + 65536;          // 32*12*128 roped q
  float* kw   = qw + 49152;          // 32*2*128  roped new k
  float* vw   = kw + 8192;           // 32*2*128  new v
  float* attn = vw + 8192;           // 32*12*128 attention output

  qkv_proj_kernel<<<dim3(128), dim3(32), 0, stream>>>(x, Wq, bq, Wk, bk, Wv,
                                                      bv, y);
  rope_kernel<<<dim3(32), dim3(128), 0, stream>>>(y, cosb, sinb, qw, kw, vw);
  attn_kernel<<<dim3(64), dim3(128), ATTN_LDS_BYTES, stream>>>(
      qw, kw, vw, k_cache, v_cache, block_table, context_lens, attn);
  out_proj_kernel<<<dim3(96), dim3(32), 0, stream>>>(attn, Wo, (float*)d_out);
}